// NUFFT_24111946400481
// MI455X (gfx1250) — compile-verified
//
#include <hip/hip_runtime.h>
#include <math.h>

// ---------------------------------------------------------------- constants
#define JTAPS   6
#define ALPHA_F 14.04f          // 2.34 * J
#define NH      320
#define KGRID   640             // 2x oversampled grid
#define NPH     5
#define NBATCH  16              // ch(8) * d(2) images per phase
#define KLEN    16000
#define PI_F    3.14159265358979323846f

typedef __attribute__((ext_vector_type(2))) float v2f;
typedef __attribute__((ext_vector_type(8))) float v8f;
typedef __attribute__((ext_vector_type(4))) unsigned int v4u;
typedef __attribute__((ext_vector_type(4))) int v4i;
typedef __attribute__((ext_vector_type(8))) int v8i;

// ------------------------------------------------------------- WMMA helper
// D(16x16 f32) = A(16x4 f32) x B(4x16 f32) + C   — native CDNA5 f32 matrix op.
__device__ __forceinline__ v8f wmma4(v2f a, v2f b, v8f c) {
    return __builtin_amdgcn_wmma_f32_16x16x4_f32(
        false, a, false, b, (short)0, c, false, false);
}

// ------------------------------------------------------------- Bessel I0
__device__ __forceinline__ float bessel_i0f(float x) {
    float ax = fabsf(x);
    if (ax < 3.75f) {
        float t = x / 3.75f; t *= t;
        return 1.0f + t*(3.5156229f + t*(3.0899424f + t*(1.2067492f +
               t*(0.2659732f + t*(0.0360768f + t*0.0045813f)))));
    } else {
        float t = 3.75f / ax;
        float p = 0.39894228f + t*(0.01328592f + t*(0.00225319f +
                  t*(-0.00157565f + t*(0.00916281f + t*(-0.02057706f +
                  t*(0.02635537f + t*(-0.01647633f + t*0.00392377f)))))));
        return expf(ax) * rsqrtf(ax) * p;
    }
}

// 1 / FT(KB kernel) at centered pixel n (apodization correction)
__device__ __forceinline__ float kb_invft(int n, float i0a) {
    float u  = (float)(n - NH/2) * (1.0f / (float)KGRID);
    float pj = PI_F * (float)JTAPS * u;
    float z2 = pj*pj - ALPHA_F*ALPHA_F;
    float z  = fmaxf(sqrtf(fabsf(z2)), 1e-6f);
    float core = (z2 > 0.0f) ? (sinf(z)/z) : (sinhf(z)/z);
    float ft = core * ((float)JTAPS / i0a);
    return 1.0f / ft;
}

// ------------------------------------------------------------- TDM helper
// Async DMA of a rows x cols f32 tile (row-major, stride=cols) into LDS,
// with the TDM LDS-pad feature inserting 1 DWORD every 128 DWORDs
// (pad_interval=6 -> 128 dw, pad_amount=0 -> 1 dw). Effective LDS pitch
// for a 640-dword row becomes 645 (odd mod 64 -> bank-conflict-free
// column access).  D# bit layout per CDNA5 ISA 8.3-8.5.
// This toolchain declares the 6-arg builtin:
//   (uint32x4 g0, int32x8 g1, int32x4 g2, int32x4 g3, int32x8 g4, i32 cpol)
#define XPITCH 645
__device__ __forceinline__ int xidx(int r, int c) {   // LDS index incl. TDM pads
    return r * XPITCH + c + (c >> 7);
}

#if __has_builtin(__builtin_amdgcn_tensor_load_to_lds)
#define HAVE_TDM 1
__device__ __forceinline__ void tdm_load_tile_f32(const float* gaddr,
                                                  unsigned ldsByteOff,
                                                  int rows, int cols) {
    unsigned long long ga = (unsigned long long)(size_t)gaddr;
    v4u g0;
    g0[0] = 1u;                                        // count=1 (valid user D#)
    g0[1] = ldsByteOff;                                // lds_addr
    g0[2] = (unsigned)(ga & 0xFFFFFFFFu);              // global_addr[31:0]
    g0[3] = (unsigned)((ga >> 32) & 0x1FFFFFFu)        // global_addr[56:32]
          | (2u << 30);                                // type=2 ("image")
    v8i g1;
    g1[0] = (2 << 16)                                  // data_size = 4B
          | (1 << 20)                                  // pad_enable
          | (6 << 22)                                  // pad_interval: 128 dw
          | (0 << 25);                                 // pad_amount: 1 dw
    g1[1] = (cols & 0xFFFF) << 16;                     // tensor_dim0 [15:0]
    g1[2] = ((cols >> 16) & 0xFFFF)                    // tensor_dim0 [31:16]
          | ((rows & 0xFFFF) << 16);                   // tensor_dim1 [15:0]
    g1[3] = ((rows >> 16) & 0xFFFF)                    // tensor_dim1 [31:16]
          | ((cols & 0xFFFF) << 16);                   // tile_dim0
    g1[4] = rows & 0xFFFF;                             // tile_dim1 (tile_dim2=0)
    g1[5] = cols;                                      // tensor_dim0_stride[31:0]
    g1[6] = 0;
    g1[7] = 0;
    v4i gz4 = {0, 0, 0, 0};                            // 2-D tensor: groups 2/3 off
    v8i gz8 = {0, 0, 0, 0, 0, 0, 0, 0};
    __builtin_amdgcn_tensor_load_to_lds(g0, g1, gz4, gz4, gz8, 0);
}
#else
#define HAVE_TDM 0
#endif

// ================================================================ kernel A
// apodize + zero-pad + ortho normalization (1/640 total for both FFT passes)
__global__ void nufft_apodize_pad(const float* __restrict__ image,
                                  float* __restrict__ dst, int ph) {
    const size_t plane = (size_t)KGRID * KGRID;
    size_t gid = (size_t)blockIdx.x * blockDim.x + threadIdx.x;
    if (gid >= (size_t)NBATCH * plane) return;
    int c  = (int)(gid % KGRID);
    int r  = (int)((gid / KGRID) % KGRID);
    int im = (int)(gid / plane);

    float re = 0.0f, iv = 0.0f;
    if (r < NH && c < NH) {
        float i0a = bessel_i0f(ALPHA_F);
        float s = kb_invft(r, i0a) * kb_invft(c, i0a) * (1.0f / (float)KGRID);
        size_t src = ((((size_t)ph * NBATCH + im) * NH + r) * NH + c) * 2;
        re = image[src]     * s;
        iv = image[src + 1] * s;
    }
    size_t o = (size_t)im * 2 * plane + (size_t)r * KGRID + c;
    dst[o]         = re;   // real plane
    dst[o + plane] = iv;   // imag plane
}

// ================================================================ kernel B
// One 640-pt DFT pass over all rows, written transposed: dst[k][row].
// Applying it twice yields the full 2-D FFT in original orientation.
// 640 = 16*40 four-step; both stages run on v_wmma_f32_16x16x4_f32,
// batching 16 rows as the GEMM N dimension. Wave32 fragment layouts
// follow CDNA5 ISA 7.12.2. Row staging uses the Tensor Data Mover.
#define DPITCH 20
#define TPITCH 44
#define XFLOATS (16 * XPITCH)               // 10320 per component plane

__global__ void nufft_fft640_pass(const float* __restrict__ src,
                                  float* __restrict__ dst) {
    extern __shared__ float sm[];
    float* xr = sm;                       // [16][645 incl. TDM pads]
    float* xi = xr + XFLOATS;
    float* Hr = xi + XFLOATS;             // [16][40][16]  (k1, n2, row)
    float* Hi = Hr + 16 * 40 * 16;
    float* Dr = Hi + 16 * 40 * 16;        // DFT16 [16][DPITCH]
    float* Di = Dr + 16 * DPITCH;
    float* Tr = Di + 16 * DPITCH;         // DFT40^T padded [48][TPITCH]
    float* Ti = Tr + 48 * TPITCH;
    float* Wr = Ti + 48 * TPITCH;         // twiddle W640^{m*n2} [16][40]
    float* Wi = Wr + 16 * 40;

    const int img  = blockIdx.x & 15;
    const int rb   = blockIdx.x >> 4;     // 40 row-blocks
    const int row0 = rb * 16;
    const int tid  = threadIdx.x;         // 128 threads = 4 waves
    const int lane = tid & 31;
    const int wave = tid >> 5;
    const int hi   = lane >> 4;           // lane half (ISA A/B frag layout)
    const int ln   = lane & 15;

    const size_t plane = (size_t)KGRID * KGRID;
    const float* sR = src + (size_t)img * 2 * plane;
    const float* sI = sR + plane;
    float* dR = dst + (size_t)img * 2 * plane;
    float* dI = dR + plane;

    // ---- async DMA of the 16-row tile into LDS (re + im planes) ---------
#if HAVE_TDM
    if (wave == 0) {
        tdm_load_tile_f32(sR + (size_t)row0 * KGRID, 0u,               16, KGRID);
        tdm_load_tile_f32(sI + (size_t)row0 * KGRID, XFLOATS * 4u,     16, KGRID);
    }
#endif

    // ---- build twiddle tables (overlaps the TDM transfer) ---------------
    for (int i = tid; i < 16 * 16; i += 128) {
        int m = i >> 4, n = i & 15;
        float s, c; sincosf(-2.0f * PI_F * (float)(m * n) / 16.0f, &s, &c);
        Dr[m * DPITCH + n] = c; Di[m * DPITCH + n] = s;
    }
    for (int i = tid; i < 48 * TPITCH; i += 128) {
        int k2 = i / TPITCH, n2 = i % TPITCH;
        float cr = 0.0f, ci = 0.0f;
        if (k2 < 40 && n2 < 40) {
            float s, c; sincosf(-2.0f * PI_F * (float)(k2 * n2) / 40.0f, &s, &c);
            cr = c; ci = s;
        }
        Tr[i] = cr; Ti[i] = ci;
    }
    for (int i = tid; i < 16 * 40; i += 128) {
        int m = i / 40, n2 = i % 40;
        float s, c; sincosf(-2.0f * PI_F * (float)(m * n2) / (float)KGRID, &s, &c);
        Wr[i] = c; Wi[i] = s;
    }

#if HAVE_TDM
    if (wave == 0) __builtin_amdgcn_s_wait_tensorcnt(0);
#else
    for (int i = tid; i < 16 * KGRID; i += 128) {   // fallback staging
        int r = i / KGRID, c = i % KGRID;
        xr[xidx(r, c)] = sR[(size_t)(row0 + r) * KGRID + c];
        xi[xidx(r, c)] = sI[(size_t)(row0 + r) * KGRID + c];
    }
#endif
    __syncthreads();

    const v8f vz = {0.f,0.f,0.f,0.f,0.f,0.f,0.f,0.f};

    // ---- step 1: G = D16 x X_{n2}  (16x16x16, complex), then twiddle ----
    for (int n2 = wave; n2 < 40; n2 += 4) {
        v8f arr = vz, aii = vz, ari = vz, air = vz;
        for (int c4 = 0; c4 < 4; ++c4) {
            int kb = 4 * c4 + 2 * hi;
            v2f Ar, Ai, Br, Bi;
            Ar.x = Dr[ln * DPITCH + kb];  Ar.y = Dr[ln * DPITCH + kb + 1];
            Ai.x = Di[ln * DPITCH + kb];  Ai.y = Di[ln * DPITCH + kb + 1];
            int c0 = 40 * kb + n2;                  // x[row][40*n1+n2]
            int c1 = c0 + 40;
            Br.x = xr[xidx(ln, c0)]; Br.y = xr[xidx(ln, c1)];
            Bi.x = xi[xidx(ln, c0)]; Bi.y = xi[xidx(ln, c1)];
            arr = wmma4(Ar, Br, arr); aii = wmma4(Ai, Bi, aii);
            ari = wmma4(Ar, Bi, ari); air = wmma4(Ai, Br, air);
        }
        for (int r = 0; r < 8; ++r) {
            int m = r + 8 * hi;                     // k1
            float gr = arr[r] - aii[r];
            float gi = ari[r] + air[r];
            float wc = Wr[m * 40 + n2], ws = Wi[m * 40 + n2];
            Hr[(m * 40 + n2) * 16 + ln] = gr * wc - gi * ws;
            Hi[(m * 40 + n2) * 16 + ln] = gr * ws + gi * wc;
        }
    }
    __syncthreads();

    // ---- step 3: Out[k2-tile][rows] = T40^T x H_{k1}  (48x40x16, complex)
    for (int t = wave; t < 48; t += 4) {
        int k1 = t / 3, mt = t % 3;
        v8f arr = vz, aii = vz, ari = vz, air = vz;
        for (int c4 = 0; c4 < 10; ++c4) {           // K = 40 exact
            int kb = 4 * c4 + 2 * hi;
            int k2g = mt * 16 + ln;                 // padded rows >=40 are zero
            v2f Ar, Ai, Br, Bi;
            Ar.x = Tr[k2g * TPITCH + kb];  Ar.y = Tr[k2g * TPITCH + kb + 1];
            Ai.x = Ti[k2g * TPITCH + kb];  Ai.y = Ti[k2g * TPITCH + kb + 1];
            Br.x = Hr[(k1 * 40 + kb) * 16 + ln];
            Br.y = Hr[(k1 * 40 + kb + 1) * 16 + ln];
            Bi.x = Hi[(k1 * 40 + kb) * 16 + ln];
            Bi.y = Hi[(k1 * 40 + kb + 1) * 16 + ln];
            arr = wmma4(Ar, Br, arr); aii = wmma4(Ai, Bi, aii);
            ari = wmma4(Ar, Bi, ari); air = wmma4(Ai, Br, air);
        }
        for (int r = 0; r < 8; ++r) {
            int m  = r + 8 * hi;
            int k2 = mt * 16 + m;
            if (k2 < 40) {
                int k = k1 + 16 * k2;
                dR[(size_t)k * KGRID + row0 + ln] = arr[r] - aii[r];
                dI[(size_t)k * KGRID + row0 + ln] = ari[r] + air[r];
            }
        }
    }
}

// ================================================================ kernel C
// Kaiser-Bessel 6x6 gather at non-uniform points + recentering phase.
__device__ __forceinline__ void kb_taps(float om, float inv_i0a,
                                        float* w, int* idx) {
    float t = om * ((float)KGRID / (2.0f * PI_F));
    float base = floorf(t - 0.5f * (float)JTAPS);
    #pragma unroll
    for (int j = 0; j < JTAPS; ++j) {
        float kf = base + (float)(j + 1);
        float u  = t - kf;
        float arg = fmaxf(1.0f - (u / 3.0f) * (u / 3.0f), 0.0f);
        w[j] = bessel_i0f(ALPHA_F * sqrtf(arg)) * inv_i0a;
        int k = (int)kf;
        idx[j] = ((k % KGRID) + KGRID) % KGRID;
    }
}

__global__ void nufft_interp(const float* __restrict__ grid,
                             const float* __restrict__ traj,
                             float* __restrict__ out, int ph) {
    int gid = blockIdx.x * blockDim.x + threadIdx.x;
    if (gid >= KLEN * NBATCH) return;
    int im = gid & 15;
    int l  = gid >> 4;

    float om0 = traj[((size_t)ph * 2 + 0) * KLEN + l];
    float om1 = traj[((size_t)ph * 2 + 1) * KLEN + l];

    float inv_i0a = 1.0f / bessel_i0f(ALPHA_F);
    float wh[JTAPS], ww[JTAPS]; int ih[JTAPS], iw[JTAPS];
    kb_taps(om0, inv_i0a, wh, ih);
    kb_taps(om1, inv_i0a, ww, iw);

    const size_t plane = (size_t)KGRID * KGRID;
    const float* gR = grid + (size_t)im * 2 * plane;
    const float* gI = gR + plane;

    #pragma unroll
    for (int i = 0; i < JTAPS; ++i)        // global_prefetch_b8 on tap rows
        __builtin_prefetch(&gR[(size_t)ih[i] * KGRID + iw[0]], 0, 3);

    float ar = 0.0f, ai = 0.0f;
    #pragma unroll
    for (int i = 0; i < JTAPS; ++i) {
        size_t rowo = (size_t)ih[i] * KGRID;
        #pragma unroll
        for (int j = 0; j < JTAPS; ++j) {
            size_t o = rowo + iw[j];
            float w = wh[i] * ww[j];
            ar += gR[o] * w;
            ai += gI[o] * w;
        }
    }
    float ps, pc;
    sincosf((float)(NH / 2) * (om0 + om1), &ps, &pc);
    size_t o = (((size_t)ph * NBATCH + im) * KLEN + l) * 2;
    out[o]     = ar * pc - ai * ps;
    out[o + 1] = ar * ps + ai * pc;
}

// ================================================================ launch
extern "C" void kernel_launch(void* const* d_in, const int* in_sizes, int n_in,
                              void* d_out, int out_size, void* d_ws, size_t ws_size,
                              hipStream_t stream) {
    const float* image = (const float*)d_in[0];  // (1,5,8,2,320,320,2)
    const float* traj  = (const float*)d_in[1];  // (5,2,16000)
    float* out  = (float*)d_out;                 // (1,5,8,2,16000,2)

    const size_t plane = (size_t)KGRID * KGRID;
    float* bufA = (float*)d_ws;                      // 16 imgs * 2 comps * 640^2
    float* bufB = bufA + (size_t)NBATCH * 2 * plane; // second ping-pong buffer

    constexpr size_t ldsFloats =
        2 * XFLOATS + 2 * (16 * 40 * 16) + 2 * (16 * DPITCH) +
        2 * (48 * TPITCH) + 2 * (16 * 40);
    constexpr size_t ldsBytes = ldsFloats * sizeof(float);   // ~185 KB (<320 KB)

    const int apodThreads = 256;
    const int apodBlocks  = (int)(((size_t)NBATCH * plane + apodThreads - 1) / apodThreads);
    const int interpBlocks = (KLEN * NBATCH + 255) / 256;

    for (int ph = 0; ph < NPH; ++ph) {   // per-phase: 52 MB grid stays in L2
        nufft_apodize_pad<<<apodBlocks, apodThreads, 0, stream>>>(image, bufA, ph);
        nufft_fft640_pass<<<40 * NBATCH, 128, ldsBytes, stream>>>(bufA, bufB);
        nufft_fft640_pass<<<40 * NBATCH, 128, ldsBytes, stream>>>(bufB, bufA);
        nufft_interp<<<interpBlocks, 256, 0, stream>>>(bufA, traj, out, ph);
    }
}